// UnConv_31250182046036
// MI455X (gfx1250) — compile-verified
//
#include <hip/hip_runtime.h>
#include <hip/hip_bf16.h>
#include <stdint.h>

// UnConv (K=2,S=2, buggy p=K): pure permutation scatter of each 64x64 plane
// into a 128x128 plane with modulo wrap. Output-centric closed-form gather:
//   oy odd  (row 127 = wrapped i=0): odd ox<=125 carry x[((oy+1)/2)&63][(ox+1)/2]
//   oy even, ox==127 (row 126 wraps): carry x[((oy/2)+1)&63][0]
//   else 0
// Zero FLOPs; 64 MiB read + 256 MiB write, ~14 us HBM floor @ 23.3 TB/s.
//
// Each thread produces a 4-row x 4-col output tile (4 float4): row parity is
// compile-time after unrolling, so the kernel is branch-free (cndmask only).
// CDNA5 path: results staged in LDS ([4][256] float4 -> conflict-free b128),
// written with GLOBAL_STORE_ASYNC_FROM_LDS_B128 (ASYNCcnt engine) after a
// wave-local s_wait_dscnt 0. S_ENDPGM's implicit wait-idle drains ASYNCcnt.

__global__ __launch_bounds__(256) void unconv_scatter_kernel(
    const float* __restrict__ x, float* __restrict__ y, int nthreads) {
  int gid = blockIdx.x * 256 + threadIdx.x;
  if (gid >= nthreads) return;

  int p     = gid & 1023;       // thread within one plane (32 rowblocks x 32)
  int plane = gid >> 10;        // (b*C + c) plane index
  int rb    = p >> 5;           // row block 0..31  (rows 4rb .. 4rb+3)
  int ox0   = (p & 31) << 2;    // output col of lane .x (multiple of 4)

  const float* __restrict__ xp = x + ((size_t)plane << 12);  // 64x64 plane

  const bool interior = ox0 < 124;   // ox0 == 124 is the wrapped col-127 slot
  const int  halfcol  = ox0 >> 1;
  const int  iA = 2 * rb + 1;        // source row for oy = 4rb (col127) & 4rb+1
  const int  iB = (2 * rb + 2) & 63; // source row for oy = 4rb+2 (col127) & 4rb+3

  float4 v[4];
  { // r = 0: even row oy = 4rb -> only wrapped col-127 element
    float t = xp[iA << 6];
    v[0] = make_float4(0.f, 0.f, 0.f, interior ? 0.f : t);
  }
  { // r = 1: odd row oy = 4rb+1, source row iA
    int base = (iA << 6) + halfcol;
    float a = xp[base + 1];                      // ox = ox0+1 (1..125)
    float b = xp[base + (interior ? 2 : 1)];     // clamped index, no OOB
    v[1] = make_float4(0.f, a, 0.f, interior ? b : 0.f);
  }
  { // r = 2: even row oy = 4rb+2 -> only wrapped col-127 element
    float t = xp[iB << 6];
    v[2] = make_float4(0.f, 0.f, 0.f, interior ? 0.f : t);
  }
  { // r = 3: odd row oy = 4rb+3, source row iB
    int base = (iB << 6) + halfcol;
    float a = xp[base + 1];
    float b = xp[base + (interior ? 2 : 1)];
    v[3] = make_float4(0.f, a, 0.f, interior ? b : 0.f);
  }

  __shared__ float4 tile[4][256];   // 16 KiB; lane stride 16B -> no conflicts
  const int tid = threadIdx.x;
#pragma unroll
  for (int r = 0; r < 4; ++r) tile[r][tid] = v[r];

  // Output base (floats): plane*16384 + (4rb)*128 + ox0
  float* yb = y + (((size_t)plane << 14) + ((size_t)rb << 9) + (size_t)ox0);

  asm volatile("s_wait_dscnt 0" ::: "memory");
#pragma unroll
  for (int r = 0; r < 4; ++r) {
    unsigned long long ga = (unsigned long long)(size_t)(yb + ((size_t)r << 7));
    uint32_t lds_off = (uint32_t)(size_t)&tile[r][tid];
    asm volatile("global_store_async_from_lds_b128 %0, %1, off"
                 :: "v"(ga), "v"(lds_off)
                 : "memory");
  }
  // Implicit S_WAIT_IDLE at S_ENDPGM drains ASYNCcnt.
}

extern "C" void kernel_launch(void* const* d_in, const int* in_sizes, int n_in,
                              void* d_out, int out_size, void* d_ws, size_t ws_size,
                              hipStream_t stream) {
  const float* x = (const float*)d_in[0];   // (16,256,64,64) f32
  float*       y = (float*)d_out;           // (16,256,128,128) f32

  int nthreads = out_size >> 4;             // 16 output floats per thread
  int blocks   = (nthreads + 255) / 256;    // 16384 blocks of 256
  unconv_scatter_kernel<<<blocks, 256, 0, stream>>>(x, y, nthreads);
}